// GAT_12876311953930
// MI455X (gfx1250) — compile-verified
//
#include <hip/hip_runtime.h>
#include <math.h>

typedef __attribute__((ext_vector_type(2))) float v2f;
typedef __attribute__((ext_vector_type(8))) float v8f;

#define NEG_SLOPE 0.2f

// ---------------------------------------------------------------------------
// float atomic max via monotone int/uint mapping (init memory to -inf)
// ---------------------------------------------------------------------------
__device__ __forceinline__ void atomicMaxF32(float* addr, float val) {
    if (val >= 0.0f) {
        atomicMax((int*)addr, __float_as_int(val));
    } else {
        atomicMin((unsigned int*)addr, __float_as_uint(val));
    }
}

// CDNA5 async global->LDS copy (16B), tracked by ASYNCcnt.
__device__ __forceinline__ void async_copy_b128(unsigned lds_off, const float* gaddr) {
    asm volatile("global_load_async_to_lds_b128 %0, %1, off"
                 :: "v"(lds_off), "v"(gaddr)
                 : "memory");
}
__device__ __forceinline__ void wait_asynccnt0() {
    asm volatile("s_wait_asynccnt 0x0" ::: "memory");
}

__global__ void fill_f32(float* __restrict__ p, float v, int n) {
    int i = blockIdx.x * blockDim.x + threadIdx.x;
    if (i < n) p[i] = v;
}

// ---------------------------------------------------------------------------
// Y[N,F] = X[N,K] * W[K,F], fp32 WMMA 16x16x4.
// Block = 256 threads = 8 waves covering `rowsPerBlock` consecutive rows x
// all F columns. Both the contiguous X row-stripe AND the whole W matrix are
// staged into LDS via async global->LDS b128 copies (ASYNCcnt), so the K-loop
// feeds every WMMA operand with 32-bit-addressed ds_loads only.
// LDS: rowsPerBlock*K + K*F floats (72 KB for F=128, 48 KB for F=64).
// ---------------------------------------------------------------------------
__global__ __launch_bounds__(256)
void gemm_wmma_f32_lds(const float* __restrict__ X, const float* __restrict__ W,
                       float* __restrict__ Y, int N, int K, int F,
                       int rowsPerBlock) {
    extern __shared__ float sh[];
    float* shXp = sh;                            // rowsPerBlock x K
    float* shW  = sh + (size_t)rowsPerBlock * K; // K x F
    const int tilesPerRow = F >> 4;              // 8 (F=128) or 4 (F=64)
    const int rowBase0 = blockIdx.x * rowsPerBlock;

    // ---- stage X stripe + full W into LDS (async copies) ----
    const float* gsrc = X + (size_t)rowBase0 * K;
    const int vecsX = (rowsPerBlock * K) >> 2;   // float4 count
    const int vecsW = (K * F) >> 2;
    const int total = vecsX + vecsW;
    for (int v4 = threadIdx.x; v4 < total; v4 += blockDim.x) {
        if (v4 < vecsX) {
            unsigned lds_off = (unsigned)(uintptr_t)(&shXp[v4 << 2]);
            async_copy_b128(lds_off, gsrc + ((size_t)v4 << 2));
        } else {
            int w4 = v4 - vecsX;
            unsigned lds_off = (unsigned)(uintptr_t)(&shW[w4 << 2]);
            async_copy_b128(lds_off, W + ((size_t)w4 << 2));
        }
    }
    wait_asynccnt0();        // this wave's async writes to LDS are complete
    __syncthreads();         // all waves' stages visible

    const int wave = threadIdx.x >> 5;
    const int lane = threadIdx.x & 31;
    const int rowStripe = wave / tilesPerRow;    // 0..(8/tilesPerRow - 1)
    const int colTile   = wave - rowStripe * tilesPerRow;
    const int rowLocal  = rowStripe << 4;
    const int colBase   = colTile << 4;
    const int m    = lane & 15;                  // A row / B,C,D column
    const int half = lane >> 4;                  // K-pair select, C/D row group

    v8f c = {};
    const float* arow = &shXp[(size_t)(rowLocal + m) * K];
    const float* bcol = &shW[colBase + m];
    for (int kb = 0; kb < K; kb += 4) {
        int k0 = kb + 2 * half;
        v2f a, b;
        a.x = arow[k0];                          // ds_load
        a.y = arow[k0 + 1];
        b.x = bcol[(size_t)k0 * F];              // ds_load
        b.y = bcol[(size_t)(k0 + 1) * F];
        c = __builtin_amdgcn_wmma_f32_16x16x4_f32(
                /*neg_a=*/false, a, /*neg_b=*/false, b,
                /*c_mod=*/(short)0, c, /*reuse_a=*/false, /*reuse_b=*/false);
    }
    const int rowG = rowBase0 + rowLocal;
#pragma unroll
    for (int v = 0; v < 8; ++v) {
        Y[(size_t)(rowG + v + 8 * half) * F + colBase + m] = c[v];
    }
}

// ---------------------------------------------------------------------------
// al_s[n,h] = dot(xp[n, h*C:(h+1)*C], a_src[h]);  al_d likewise (float4 path)
// ---------------------------------------------------------------------------
__global__ void attn_coef(const float* __restrict__ xp,
                          const float* __restrict__ a_src,
                          const float* __restrict__ a_dst,
                          float* __restrict__ als, float* __restrict__ ald,
                          int N, int H, int C) {
    int i = blockIdx.x * blockDim.x + threadIdx.x;
    if (i >= N * H) return;
    int n = i / H, h = i - n * H;
    const float4* xr = (const float4*)(xp + (size_t)n * H * C + (size_t)h * C);
    const float4* as = (const float4*)(a_src + (size_t)h * C);
    const float4* ad = (const float4*)(a_dst + (size_t)h * C);
    float s1 = 0.0f, s2 = 0.0f;
    int c4 = C >> 2;
    for (int c = 0; c < c4; ++c) {
        float4 v = xr[c];
        float4 a = as[c];
        float4 d = ad[c];
        s1 += v.x * a.x + v.y * a.y + v.z * a.z + v.w * a.w;
        s2 += v.x * d.x + v.y * d.y + v.z * d.z + v.w * d.w;
    }
    als[i] = s1;
    ald[i] = s2;
}

__device__ __forceinline__ void get_edge(const int* __restrict__ ei, int EB,
                                         int e, int& s, int& d) {
    if (e < EB) { s = ei[e]; d = ei[EB + e]; }
    else        { s = e - EB; d = e - EB; }   // appended self-loops
}

// pass 1: e = leaky_relu(al_s[src]+al_d[dst]); store; segment max into m[dst]
__global__ void edge_pass1(const int* __restrict__ ei, int EB, int ET,
                           const float* __restrict__ als,
                           const float* __restrict__ ald,
                           float* __restrict__ ew, float* __restrict__ mbuf,
                           int H) {
    int e = blockIdx.x * blockDim.x + threadIdx.x;
    if (e >= ET) return;
    int s, d;
    get_edge(ei, EB, e, s, d);
    for (int h = 0; h < H; ++h) {
        float v = als[(size_t)s * H + h] + ald[(size_t)d * H + h];
        v = v > 0.0f ? v : NEG_SLOPE * v;
        ew[(size_t)e * H + h] = v;
        atomicMaxF32(&mbuf[(size_t)d * H + h], v);
    }
}

// pass 2: ex = exp(e - m[dst]); store; segment sum into s[dst]
__global__ void edge_pass2(const int* __restrict__ ei, int EB, int ET,
                           float* __restrict__ ew,
                           const float* __restrict__ mbuf,
                           float* __restrict__ sbuf, int H) {
    int e = blockIdx.x * blockDim.x + threadIdx.x;
    if (e >= ET) return;
    int s, d;
    get_edge(ei, EB, e, s, d);
    for (int h = 0; h < H; ++h) {
        float ex = expf(ew[(size_t)e * H + h] - mbuf[(size_t)d * H + h]);
        ew[(size_t)e * H + h] = ex;
        atomicAdd(&sbuf[(size_t)d * H + h], ex);
    }
}

// pass 3: agg[dst] += xp[src] * alpha  (32 lanes/edge, vectorized L2 gather)
__global__ __launch_bounds__(256)
void edge_pass3(const int* __restrict__ ei, int EB, int ET,
                const float* __restrict__ xp, const float* __restrict__ ew,
                const float* __restrict__ sbuf, float* __restrict__ agg,
                int H, int F) {
    int gid  = blockIdx.x * blockDim.x + threadIdx.x;
    int e    = gid >> 5;
    int lane = gid & 31;
    if (e >= ET) return;
    int s, d;
    get_edge(ei, EB, e, s, d);
    int fpl = F >> 5;            // 4 (F=128) or 2 (F=64); stays inside one head
    int f0  = lane * fpl;
    int C   = F / H;
    int h   = f0 / C;
    float alpha = ew[(size_t)e * H + h] /
                  (sbuf[(size_t)d * H + h] + 1e-16f);
    const float* xs = xp + (size_t)s * F + f0;
    float*       ag = agg + (size_t)d * F + f0;
    if (fpl == 4) {
        float4 v = *(const float4*)xs;         // global_load_b128 (L2 hit)
        atomicAdd(&ag[0], v.x * alpha);
        atomicAdd(&ag[1], v.y * alpha);
        atomicAdd(&ag[2], v.z * alpha);
        atomicAdd(&ag[3], v.w * alpha);
    } else {
        float2 v = *(const float2*)xs;         // global_load_b64
        atomicAdd(&ag[0], v.x * alpha);
        atomicAdd(&ag[1], v.y * alpha);
    }
}

// out = agg + bias, optional ELU
__global__ void finalize_k(const float* __restrict__ agg,
                           const float* __restrict__ b,
                           float* __restrict__ out, int N, int F, int do_elu) {
    int i = blockIdx.x * blockDim.x + threadIdx.x;
    if (i >= N * F) return;
    int f = i % F;
    float v = agg[i] + b[f];
    if (do_elu) v = v > 0.0f ? v : expm1f(v);
    out[i] = v;
}

// ---------------------------------------------------------------------------
extern "C" void kernel_launch(void* const* d_in, const int* in_sizes, int n_in,
                              void* d_out, int out_size, void* d_ws, size_t ws_size,
                              hipStream_t stream) {
    const float* x  = (const float*)d_in[0];
    const int*   ei = (const int*)d_in[1];
    const float* W[3]    = { (const float*)d_in[2], (const float*)d_in[6],  (const float*)d_in[10] };
    const float* Asrc[3] = { (const float*)d_in[3], (const float*)d_in[7],  (const float*)d_in[11] };
    const float* Adst[3] = { (const float*)d_in[4], (const float*)d_in[8],  (const float*)d_in[12] };
    const float* B[3]    = { (const float*)d_in[5], (const float*)d_in[9],  (const float*)d_in[13] };

    const int N  = in_sizes[0] / 128;   // 100000
    const int EB = in_sizes[1] / 2;     // 1600000
    const int ET = EB + N;              // + self loops
    const int HMAX = 4;

    // workspace carve-up
    float* xp   = (float*)d_ws;          // N*128
    float* hA   = xp  + (size_t)N * 128; // N*128
    float* hB   = hA  + (size_t)N * 128; // N*128
    float* als  = hB  + (size_t)N * 128; // N*HMAX
    float* ald  = als + (size_t)N * HMAX;
    float* mbuf = ald + (size_t)N * HMAX;
    float* sbuf = mbuf + (size_t)N * HMAX;
    float* ew   = sbuf + (size_t)N * HMAX; // ET*HMAX
    (void)ws_size; (void)n_in; (void)out_size;

    const int TB = 256;
    const float* hin[3]  = { x, hA, hB };
    float*       hout[3] = { hA, hB, (float*)d_out };
    const int    Fout[3] = { 128, 128, 64 };
    const int    Hh[3]   = { 4, 4, 1 };
    const int    doElu[3] = { 1, 1, 0 };

    for (int L = 0; L < 3; ++L) {
        const int F = Fout[L], H = Hh[L], C = F / H;
        const int K = 128;

        // 1) xp = hin * W  (fp32 WMMA, A-stripe + W staged in LDS via async)
        int tilesPerRow  = F >> 4;                    // 8 or 4
        int rowsPerBlock = (8 / tilesPerRow) << 4;    // 16 or 32
        int gBlk         = N / rowsPerBlock;          // exact: 6250 / 3125
        size_t ldsBytes  = ((size_t)rowsPerBlock * K + (size_t)K * F) * sizeof(float);
        gemm_wmma_f32_lds<<<gBlk, TB, ldsBytes, stream>>>(
            hin[L], W[L], xp, N, K, F, rowsPerBlock);

        // 2) attention logits per (node, head)
        attn_coef<<<(N * H + TB - 1) / TB, TB, 0, stream>>>(
            xp, Asrc[L], Adst[L], als, ald, N, H, C);

        // 3) init segment stats + aggregation target
        fill_f32<<<(N * H + TB - 1) / TB, TB, 0, stream>>>(mbuf, -INFINITY, N * H);
        fill_f32<<<(N * H + TB - 1) / TB, TB, 0, stream>>>(sbuf, 0.0f, N * H);
        fill_f32<<<(N * F + TB - 1) / TB, TB, 0, stream>>>(hout[L], 0.0f, N * F);

        // 4) softmax over incoming edges + weighted scatter
        int eBlk = (ET + TB - 1) / TB;
        edge_pass1<<<eBlk, TB, 0, stream>>>(ei, EB, ET, als, ald, ew, mbuf, H);
        edge_pass2<<<eBlk, TB, 0, stream>>>(ei, EB, ET, ew, mbuf, sbuf, H);
        long long tot3 = (long long)ET * 32;
        edge_pass3<<<(int)((tot3 + TB - 1) / TB), TB, 0, stream>>>(
            ei, EB, ET, xp, ew, sbuf, hout[L], H, F);

        // 5) bias + ELU (in place)
        finalize_k<<<(N * F + TB - 1) / TB, TB, 0, stream>>>(
            hout[L], B[L], hout[L], N, F, doElu[L]);
    }
}